// MGATLayer_23330262352404
// MI455X (gfx1250) — compile-verified
//
#include <hip/hip_runtime.h>

typedef float v2f __attribute__((ext_vector_type(2)));
typedef float v8f __attribute__((ext_vector_type(8)));

#define F_IN  64
#define F_OUT 64
#define HEADS 4
#define HC    256   // HEADS * F_OUT

// ---- monotonic float<->uint mapping for atomic max on floats ----
__device__ __forceinline__ unsigned fmap(float f) {
  unsigned u = __float_as_uint(f);
  return (u & 0x80000000u) ? ~u : (u | 0x80000000u);
}
__device__ __forceinline__ float funmap(unsigned m) {
  unsigned u = (m & 0x80000000u) ? (m & 0x7FFFFFFFu) : ~m;
  return __uint_as_float(u);
}

// ---------------------------------------------------------------------------
// h[N,256] = x[N,64] @ W[64,256]   (fp32 WMMA, one wave per 16x16 tile)
// ---------------------------------------------------------------------------
__global__ void k_gemm_wmma(const float* __restrict__ x, const float* __restrict__ W,
                            float* __restrict__ h, int n_nodes) {
  int wave = (blockIdx.x * blockDim.x + threadIdx.x) >> 5;
  int lane = threadIdx.x & 31;
  int tileM = wave >> 4;          // 16 tiles across the 256 output columns
  int tileN = wave & 15;
  int m0 = tileM * 16;
  if (m0 >= n_nodes) return;

  int rlo = lane & 15;            // A: M index / B,C: N index
  int kh  = lane >> 4;            // A: K half (K=2*kh..2*kh+1) / C: row half
  int r   = m0 + rlo; if (r >= n_nodes) r = n_nodes - 1;

  const float* xr = x + (size_t)r * F_IN;
  const float* wc = W + tileN * 16 + rlo;

  v8f acc = {0.f, 0.f, 0.f, 0.f, 0.f, 0.f, 0.f, 0.f};
#pragma unroll
  for (int k0 = 0; k0 < F_IN; k0 += 4) {
    int k = k0 + 2 * kh;
    v2f a; a.x = xr[k];                    a.y = xr[k + 1];
    v2f b; b.x = wc[(size_t)k * HC];       b.y = wc[(size_t)(k + 1) * HC];
    acc = __builtin_amdgcn_wmma_f32_16x16x4_f32(
        /*neg_a=*/false, a, /*neg_b=*/false, b,
        /*c_mod=*/(short)0, acc, /*reuse_a=*/false, /*reuse_b=*/false);
  }
#pragma unroll
  for (int v = 0; v < 8; ++v) {
    int rr = m0 + v + 8 * kh;              // C layout: VGPR v -> rows v / v+8
    if (rr < n_nodes) h[(size_t)rr * HC + tileN * 16 + rlo] = acc[v];
  }
}

// ---------------------------------------------------------------------------
// a_s[n,h] = <h[n,h,:], att_src[h,:]> ; a_d likewise.  One wave per node.
// ---------------------------------------------------------------------------
__global__ void k_att(const float* __restrict__ h, const float* __restrict__ att_s,
                      const float* __restrict__ att_d, float* __restrict__ a_s,
                      float* __restrict__ a_d, int n_nodes) {
  int wave = (blockIdx.x * blockDim.x + threadIdx.x) >> 5;
  int lane = threadIdx.x & 31;
  if (wave >= n_nodes) return;
  const float* hn = h + (size_t)wave * HC;
#pragma unroll
  for (int hd = 0; hd < HEADS; ++hd) {
    float ps = 0.f, pd = 0.f;
#pragma unroll
    for (int j = 0; j < 2; ++j) {
      int c = lane + j * 32;
      float hv = hn[hd * F_OUT + c];
      ps += hv * att_s[hd * F_OUT + c];
      pd += hv * att_d[hd * F_OUT + c];
    }
    for (int off = 16; off > 0; off >>= 1) {
      ps += __shfl_down(ps, off, 32);
      pd += __shfl_down(pd, off, 32);
    }
    if (lane == 0) { a_s[wave * HEADS + hd] = ps; a_d[wave * HEADS + hd] = pd; }
  }
}

__global__ void k_clear(unsigned* __restrict__ m, float* __restrict__ denom, int n) {
  int i = blockIdx.x * blockDim.x + threadIdx.x;
  if (i < n) { m[i] = 0u; denom[i] = 0.f; }
}

__global__ void k_init_out(float* __restrict__ out, const float* __restrict__ bias_intra,
                           const float* __restrict__ bias_inter, int total) {
  int i = blockIdx.x * blockDim.x + threadIdx.x;
  if (i >= total) return;
  int c = i & (F_OUT - 1);
  out[i] = bias_intra[c] + bias_intra[F_OUT + c] + bias_intra[2 * F_OUT + c] + bias_inter[c];
}

// ---------------------------------------------------------------------------
// Pass 1: e = leaky_relu(a_s[src]+a_d[dst]); atomic segment-max into mmax[dst].
// ---------------------------------------------------------------------------
__global__ void k_edge_logits(const int* __restrict__ src, const int* __restrict__ dst,
                              const float* __restrict__ a_s, const float* __restrict__ a_d,
                              const int* __restrict__ modality, int modval,
                              float* __restrict__ ebuf, unsigned* __restrict__ mmax,
                              int n_edges) {
  int e = blockIdx.x * blockDim.x + threadIdx.x;
  if (e >= n_edges) return;
  if (modality && modality[e] != modval) return;
  int s = src[e], d = dst[e];
#pragma unroll
  for (int hd = 0; hd < HEADS; ++hd) {
    float v = a_s[s * HEADS + hd] + a_d[d * HEADS + hd];
    v = v > 0.f ? v : 0.2f * v;                 // leaky relu, slope 0.2
    ebuf[(size_t)e * HEADS + hd] = v;
    atomicMax(&mmax[d * HEADS + hd], fmap(v));
  }
}

// ---------------------------------------------------------------------------
// Pass 2: p = exp(e - max); atomic segment-sum of p into denom[dst].
// ---------------------------------------------------------------------------
__global__ void k_edge_exp(const int* __restrict__ dst, const unsigned* __restrict__ mmax,
                           const int* __restrict__ modality, int modval,
                           float* __restrict__ ebuf, float* __restrict__ denom,
                           int n_edges) {
  int e = blockIdx.x * blockDim.x + threadIdx.x;
  if (e >= n_edges) return;
  if (modality && modality[e] != modval) return;
  int d = dst[e];
#pragma unroll
  for (int hd = 0; hd < HEADS; ++hd) {
    float mm = funmap(mmax[d * HEADS + hd]);
    float p  = __expf(ebuf[(size_t)e * HEADS + hd] - mm);
    ebuf[(size_t)e * HEADS + hd] = p;
    atomicAdd(&denom[d * HEADS + hd], p);
  }
}

// ---------------------------------------------------------------------------
// Pass 3: out[dst] += (1/H) * sum_h alpha_h * h[src,h,:].  One wave per edge,
// lane -> columns {lane, lane+32}; 2 fp32 atomics per lane (L2-resident out).
// ---------------------------------------------------------------------------
__global__ void k_edge_aggr(const int* __restrict__ src, const int* __restrict__ dst,
                            const float* __restrict__ ebuf, const float* __restrict__ denom,
                            const float* __restrict__ h, const int* __restrict__ modality,
                            int modval, float* __restrict__ out, int n_edges) {
  int e    = (blockIdx.x * blockDim.x + threadIdx.x) >> 5;
  int lane = threadIdx.x & 31;
  if (e >= n_edges) return;
  if (modality && modality[e] != modval) return;
  int s = src[e], d = dst[e];
  float al[HEADS];
#pragma unroll
  for (int hd = 0; hd < HEADS; ++hd) {
    float dn = denom[d * HEADS + hd];
    al[hd] = ebuf[(size_t)e * HEADS + hd] / fmaxf(dn, 1e-16f);
  }
  const float* hs = h + (size_t)s * HC;
  float acc0 = 0.f, acc1 = 0.f;
#pragma unroll
  for (int hd = 0; hd < HEADS; ++hd) {
    acc0 += al[hd] * hs[hd * F_OUT + lane];
    acc1 += al[hd] * hs[hd * F_OUT + lane + 32];
  }
  atomicAdd(&out[(size_t)d * F_OUT + lane],      0.25f * acc0);
  atomicAdd(&out[(size_t)d * F_OUT + lane + 32], 0.25f * acc1);
}

// ---------------------------------------------------------------------------
extern "C" void kernel_launch(void* const* d_in, const int* in_sizes, int n_in,
                              void* d_out, int out_size, void* d_ws, size_t ws_size,
                              hipStream_t stream) {
  const float* x             = (const float*)d_in[0];
  const int*   modality      = (const int*)  d_in[1];
  const int*   intra_src     = (const int*)  d_in[2];
  const int*   intra_dst     = (const int*)  d_in[3];
  const int*   inter_src     = (const int*)  d_in[4];
  const int*   inter_dst     = (const int*)  d_in[5];
  const float* W_intra       = (const float*)d_in[6];
  const float* att_src_intra = (const float*)d_in[7];
  const float* att_dst_intra = (const float*)d_in[8];
  const float* bias_intra    = (const float*)d_in[9];
  const float* W_inter       = (const float*)d_in[10];
  const float* att_src_inter = (const float*)d_in[11];
  const float* att_dst_inter = (const float*)d_in[12];
  const float* bias_inter    = (const float*)d_in[13];
  float* out = (float*)d_out;

  int n_nodes = in_sizes[0] / F_IN;
  int e_intra = in_sizes[1];
  int e_inter = in_sizes[4];
  int e_max   = e_intra > e_inter ? e_intra : e_inter;

  // Workspace carve-out (~123 MB for the reference sizes)
  char* ws = (char*)d_ws;
  float*    h    = (float*)ws;    ws += (size_t)n_nodes * HC    * sizeof(float);
  float*    a_s  = (float*)ws;    ws += (size_t)n_nodes * HEADS * sizeof(float);
  float*    a_d  = (float*)ws;    ws += (size_t)n_nodes * HEADS * sizeof(float);
  unsigned* mm   = (unsigned*)ws; ws += (size_t)n_nodes * HEADS * sizeof(unsigned);
  float*    dn   = (float*)ws;    ws += (size_t)n_nodes * HEADS * sizeof(float);
  float*    eb   = (float*)ws;    ws += (size_t)e_max   * HEADS * sizeof(float);

  {
    int tot = n_nodes * F_OUT;
    k_init_out<<<(tot + 255) / 256, 256, 0, stream>>>(out, bias_intra, bias_inter, tot);
  }

  for (int pass = 0; pass < 4; ++pass) {
    const float* W   = (pass < 3) ? (W_intra       + (size_t)pass * F_IN * HC) : W_inter;
    const float* ats = (pass < 3) ? (att_src_intra + (size_t)pass * HC)        : att_src_inter;
    const float* atd = (pass < 3) ? (att_dst_intra + (size_t)pass * HC)        : att_dst_inter;
    const int*  esrc = (pass < 3) ? intra_src : inter_src;
    const int*  edst = (pass < 3) ? intra_dst : inter_dst;
    const int*  emod = (pass < 3) ? modality  : nullptr;
    int ne           = (pass < 3) ? e_intra   : e_inter;
    int modval = pass;

    int tiles = ((n_nodes + 15) / 16) * (HC / 16);   // waves (one 16x16 tile each)
    k_gemm_wmma<<<(tiles + 7) / 8, 256, 0, stream>>>(x, W, h, n_nodes);
    k_att<<<(n_nodes + 7) / 8, 256, 0, stream>>>(h, ats, atd, a_s, a_d, n_nodes);

    int nh = n_nodes * HEADS;
    k_clear<<<(nh + 255) / 256, 256, 0, stream>>>(mm, dn, nh);

    k_edge_logits<<<(ne + 255) / 256, 256, 0, stream>>>(esrc, edst, a_s, a_d,
                                                        emod, modval, eb, mm, ne);
    k_edge_exp<<<(ne + 255) / 256, 256, 0, stream>>>(edst, mm, emod, modval, eb, dn, ne);
    k_edge_aggr<<<(ne + 7) / 8, 256, 0, stream>>>(esrc, edst, eb, dn, h,
                                                  emod, modval, out, ne);
  }
}